// Shift_50603304681585
// MI455X (gfx1250) — compile-verified
//
#include <hip/hip_runtime.h>
#include <hip/hip_bf16.h>

// ---------------------------------------------------------------------------
// Shift SSM layer == causal depthwise 64-tap FIR + D*u bypass, done with
// V_WMMA_F32_16X16X4_F32 (Toeplitz-as-GEMM), full fp32 precision.
// ---------------------------------------------------------------------------

typedef __attribute__((ext_vector_type(2))) float v2f;
typedef __attribute__((ext_vector_type(8))) float v8f;

#define D_MODEL   512
#define D_STATE   64
#define SEQ_L     4096
#define KTAPS     64
#define CONTRACT  80          // 64 taps + 15 intra-block shift, padded to 80 (20 x K=4)
#define TILE_T    2048        // outputs per workgroup
#define WAVE_T    256         // outputs per wave (16x16 tile)
#define NTHREADS  256         // 8 wave32s

// Kernel 1: K[h,k] = sum_s B[h,s] * C[h,s+k]  (cross-correlation over state dim)
__global__ void build_taps_kernel(const float* __restrict__ B,
                                  const float* __restrict__ C,
                                  float* __restrict__ K) {
  const int h = blockIdx.x;         // 0..511
  const int k = threadIdx.x;        // 0..63
  const float* Bh = B + h * D_STATE;
  const float* Ch = C + h * D_STATE;
  float acc = 0.0f;
  for (int s = 0; s + k < D_STATE; ++s)
    acc = __builtin_fmaf(Bh[s], Ch[s + k], acc);
  K[h * KTAPS + k] = acc;
}

// Kernel 2: y[b,h,t] = sum_k K[h,k]*u[b,h,t-k] + D[h]*u[b,h,t]
__global__ __launch_bounds__(NTHREADS)
void shift_fir_wmma_kernel(const float* __restrict__ u,
                           const float* __restrict__ K,
                           const float* __restrict__ Dvec,
                           float* __restrict__ y) {
  __shared__ float sU[TILE_T + KTAPS];     // u[t0-64 .. t0+2048), zero-padded halo
  __shared__ float sA[16 * CONTRACT];      // Toeplitz A[r][m] = K[r-m+64]
  __shared__ float sY[TILE_T];             // output staging for coalesced stores

  const int tid   = threadIdx.x;
  const int lane  = tid & 31;
  const int wave  = tid >> 5;
  const int r     = lane & 15;             // M row (A frag) / N col (B frag)
  const int half  = lane >> 4;             // 0: K={0,1}, 1: K={2,3}

  const int tileT = blockIdx.x * TILE_T;   // first output t of this workgroup
  const int h     = blockIdx.y;
  const int b     = blockIdx.z;

  const float* __restrict__ urow = u + ((size_t)b * D_MODEL + h) * SEQ_L;
  float*       __restrict__ yrow = y + ((size_t)b * D_MODEL + h) * SEQ_L;
  const float* __restrict__ Kh   = K + h * KTAPS;

  // Stage u tile (+64 halo on the left, zero-fill t<0). Coalesced b32 loads.
  for (int i = tid; i < TILE_T + KTAPS; i += NTHREADS) {
    const int t = tileT + i - KTAPS;
    sU[i] = (t >= 0) ? urow[t] : 0.0f;
  }
  // Build the 16x80 Toeplitz tap matrix once per workgroup (shared by 8 waves).
  for (int i = tid; i < 16 * CONTRACT; i += NTHREADS) {
    const int rr = i / CONTRACT;
    const int m  = i % CONTRACT;
    const int kk = rr - m + KTAPS;         // tap index
    sA[i] = (kk >= 0 && kk < KTAPS) ? Kh[kk] : 0.0f;
  }
  __syncthreads();

  // Each wave owns one 16x16 f32 tile: element (M,N) -> t = waveT0 + 16*N + M.
  const int waveT0 = wave * WAVE_T;

  v8f acc = {};
  #pragma unroll
  for (int kp = 0; kp < CONTRACT / 4; ++kp) {
    const int m0 = 4 * kp + 2 * half;      // this lane's K-slot pair {m0, m0+1}
    // A fragment: A[M=r, K] from LDS Toeplitz matrix (2 VGPRs).
    v2f afrag;
    afrag.x = sA[r * CONTRACT + m0];
    afrag.y = sA[r * CONTRACT + m0 + 1];
    // B fragment: B[K, N=r] = u[t0 + 16*N + m - 64]  ->  sU[waveT0 + 16*N + m].
    v2f bfrag;
    const int bi = waveT0 + 16 * r + m0;
    bfrag.x = sU[bi];
    bfrag.y = sU[bi + 1];
    // D = A x B + C  (f32, 16x16x4); 8 args: neg_a, A, neg_b, B, c_mod, C, reuse_a, reuse_b
    acc = __builtin_amdgcn_wmma_f32_16x16x4_f32(
        false, afrag, false, bfrag, (short)0, acc, false, false);
  }

  // Epilogue: + D[h]*u, stage to LDS. D layout: VGPR i, lane -> M=i+8*half, N=r.
  const float d = Dvec[h];
  const int obase = waveT0 + 16 * r + 8 * half;
  #pragma unroll
  for (int i = 0; i < 8; ++i) {
    sY[obase + i] = acc[i] + d * sU[obase + i + KTAPS];
  }
  __syncthreads();

  // Coalesced float4 stores: each thread writes 8 contiguous outputs.
  {
    const int i0 = tid * 8;
    const float4 v0 = *(const float4*)&sY[i0];
    const float4 v1 = *(const float4*)&sY[i0 + 4];
    *(float4*)&yrow[tileT + i0]     = v0;
    *(float4*)&yrow[tileT + i0 + 4] = v1;
  }
}

extern "C" void kernel_launch(void* const* d_in, const int* in_sizes, int n_in,
                              void* d_out, int out_size, void* d_ws, size_t ws_size,
                              hipStream_t stream) {
  (void)in_sizes; (void)n_in; (void)out_size; (void)ws_size;
  const float* u = (const float*)d_in[0];   // (8, 512, 4096)
  const float* B = (const float*)d_in[1];   // (512, 64)
  const float* C = (const float*)d_in[2];   // (1, 512, 64)
  const float* D = (const float*)d_in[3];   // (512,)
  float* y = (float*)d_out;                 // (8, 512, 4096)
  float* K = (float*)d_ws;                  // 512*64 floats = 128 KB scratch

  // 1) collapse the FFT pipeline into 64 FIR taps per channel
  build_taps_kernel<<<dim3(D_MODEL), dim3(D_STATE), 0, stream>>>(B, C, K);

  // 2) depthwise causal FIR via f32 WMMA
  dim3 grid(SEQ_L / TILE_T, D_MODEL, 8);    // (2, 512, 8)
  shift_fir_wmma_kernel<<<grid, dim3(NTHREADS), 0, stream>>>(u, K, D, y);
}